// GraphSAGE_Model_77756087927556
// MI455X (gfx1250) — compile-verified
//
#include <hip/hip_runtime.h>
#include <hip/hip_bf16.h>

typedef float v2f __attribute__((ext_vector_type(2)));
typedef float v8f __attribute__((ext_vector_type(8)));

#define HD 256
#define NT 8  // 16x16 output tiles per wave (16 rows x 128 cols)

// ---------------------------------------------------------------------------
// Tiled f32 WMMA GEMM:  out[M,256] = A0*diag? @ B0 (+ A1 @ B1) + bias, opt ReLU
// Block: 8 waves x (16 rows x 128 cols) = 128x128 output tile, grid.y = 2.
// B panel staged in LDS once per block in fragment order:
//   Bs[t][k/4][lane] = v2f{ B[(4kq+2*(lane>>4))*256 + n], B[..+256 + n] }
//   -> ds_load_b64 per fragment, lane stride 8B = conflict-free (banks 2l,2l+1)
// A fragment (16x4 f32): lane l holds row l%16, K = 2*(l/16)+{0,1}
// C/D (16x16 f32): lane l col l%16, VGPR v -> row v + 8*(l/16)
// ---------------------------------------------------------------------------
template <bool RELU, bool DUAL>
__global__ __launch_bounds__(256) void gemm_wmma(
    const float* __restrict__ A0, const float* __restrict__ B0,
    const float* __restrict__ rowscale,                 // nullable, scales A0 rows
    const float* __restrict__ A1, const float* __restrict__ B1,
    const float* __restrict__ bias,
    float* __restrict__ out, int M, int K) {
  extern __shared__ char smem[];
  v2f* Bs = (v2f*)smem;                    // [NT][K/4][32]

  const int tid = threadIdx.x;
  const int lane = tid & 31;
  const int wave = tid >> 5;
  const int m0 = (blockIdx.x * 8 + wave) * 16;
  const int n0 = blockIdx.y * (NT * 16);
  const int mr = lane & 15;
  const int kh = lane >> 4;                // K half-group (0/1)
  const bool active = (m0 < M);
  const int row = (active ? m0 : 0) + mr;  // clamp so inactive waves load safely
  const int KQ = K >> 2;

  v8f cacc[NT];
#pragma unroll
  for (int t = 0; t < NT; ++t) cacc[t] = (v8f){};

  const float* A = A0;
  const float* B = B0;
  float sc = 1.0f;
  if (rowscale) sc = rowscale[row];

  const int npass = DUAL ? 2 : 1;
  for (int pass = 0; pass < npass; ++pass) {
    // ---- stage B panel into LDS (fragment order), cooperative ----
    if (pass) __syncthreads();             // protect previous panel reads
#pragma unroll
    for (int t = 0; t < NT; ++t) {
      for (int idx = tid; idx < KQ * 32; idx += 256) {
        const int l = idx & 31;
        const int kq = idx >> 5;
        const int ka = (kq << 2) + ((l >> 4) << 1);
        const int n = n0 + (t << 4) + (l & 15);
        v2f b;
        b.x = B[ka * HD + n];
        b.y = B[(ka + 1) * HD + n];
        Bs[(t * KQ + kq) * 32 + l] = b;
      }
    }
    __syncthreads();

    // ---- MAC loop: 1 A fragment feeds 8 WMMAs ----
    for (int kq = 0; kq < KQ; ++kq) {
      const int ka = (kq << 2) + (kh << 1);
      v2f a;
      a.x = A[row * K + ka] * sc;
      a.y = A[row * K + ka + 1] * sc;
#pragma unroll
      for (int t = 0; t < NT; ++t) {
        v2f b = Bs[(t * KQ + kq) * 32 + lane];
        cacc[t] = __builtin_amdgcn_wmma_f32_16x16x4_f32(
            false, a, false, b, (short)0, cacc[t], false, false);
      }
    }
    A = A1; B = B1; sc = 1.0f;
  }

  if (active) {
#pragma unroll
    for (int t = 0; t < NT; ++t) {
      const int n = n0 + (t << 4) + mr;
      const float bn = bias ? bias[n] : 0.0f;
#pragma unroll
      for (int v = 0; v < 8; ++v) {
        float val = cacc[t][v] + bn;
        if (RELU) val = fmaxf(val, 0.0f);
        out[(size_t)(m0 + v + (kh << 3)) * HD + n] = val;
      }
    }
  }
}

// ---------------------------------------------------------------------------
// Degree / edge scatter
// ---------------------------------------------------------------------------
__global__ void deg_kernel(const int* __restrict__ dst, float* __restrict__ deg, int nE) {
  int e = blockIdx.x * blockDim.x + threadIdx.x;
  if (e < nE) atomicAdd(&deg[dst[e]], 1.0f);
}

__global__ void invdeg_kernel(float* __restrict__ deg, int n) {
  int i = blockIdx.x * blockDim.x + threadIdx.x;
  if (i < n) deg[i] = 1.0f / fmaxf(deg[i], 1.0f);
}

// one wave per edge; lane handles 8 strided columns (coalesced 1KB row moves)
__global__ __launch_bounds__(256) void scatter_add(
    const float* __restrict__ h, const int* __restrict__ src,
    const int* __restrict__ dst, float* __restrict__ agg, int nE) {
  const int lane = threadIdx.x & 31;
  int w = (blockIdx.x * blockDim.x + threadIdx.x) >> 5;
  const int nW = (gridDim.x * blockDim.x) >> 5;
  for (int e = w; e < nE; e += nW) {
    const int s = src[e] * HD;
    const int d = dst[e] * HD;
#pragma unroll
    for (int i = 0; i < 8; ++i) {
      const int col = lane + (i << 5);
      atomicAdd(&agg[d + col], h[s + col]);
    }
  }
}

// ---------------------------------------------------------------------------
// BatchNorm: column sums / sums-of-squares, finalize, apply + ReLU + residual
// ---------------------------------------------------------------------------
__global__ __launch_bounds__(256) void bn_stats(const float* __restrict__ hn,
                                                float* __restrict__ sums,
                                                float* __restrict__ sumsq, int M) {
  const int c = threadIdx.x;
  const int r0 = blockIdx.x * 512;
  const int rend = min(r0 + 512, M);
  float s = 0.0f, q = 0.0f;
  for (int r = r0; r < rend; ++r) {
    float v = hn[(size_t)r * HD + c];
    s += v;
    q += v * v;
  }
  atomicAdd(&sums[c], s);
  atomicAdd(&sumsq[c], q);
}

__global__ void bn_finalize(const float* __restrict__ sums, const float* __restrict__ sumsq,
                            const float* __restrict__ gamma, const float* __restrict__ beta,
                            float* __restrict__ scale, float* __restrict__ shift, float invM) {
  const int c = threadIdx.x;
  const float mu = sums[c] * invM;
  const float var = fmaxf(sumsq[c] * invM - mu * mu, 0.0f);
  const float sc = gamma[c] * rsqrtf(var + 1e-5f);
  scale[c] = sc;
  shift[c] = beta[c] - mu * sc;
}

__global__ __launch_bounds__(256) void bn_apply(const float* __restrict__ hn,
                                                const float* __restrict__ scale,
                                                const float* __restrict__ shift,
                                                float* __restrict__ h, int total) {
  const int i = blockIdx.x * blockDim.x + threadIdx.x;
  if (i >= total) return;
  const int c = i & (HD - 1);
  const float v = hn[i] * scale[c] + shift[c];
  h[i] += fmaxf(v, 0.0f);
}

// ---------------------------------------------------------------------------
// Global mean pool (batch is sorted -> accumulate runs, one atomic per run)
// ---------------------------------------------------------------------------
__global__ __launch_bounds__(256) void pool_kernel(const float* __restrict__ h,
                                                   const int* __restrict__ batch,
                                                   float* __restrict__ pooled,
                                                   float* __restrict__ cnt, int M) {
  const int c = threadIdx.x;
  const int r0 = blockIdx.x * 512;
  const int rend = min(r0 + 512, M);
  float acc = 0.0f;
  float n = 0.0f;
  int curg = -1;
  for (int r = r0; r < rend; ++r) {
    const int g = batch[r];
    if (g != curg) {
      if (curg >= 0) {
        atomicAdd(&pooled[curg * HD + c], acc);
        if (c == 0) atomicAdd(&cnt[curg], n);
      }
      acc = 0.0f;
      n = 0.0f;
      curg = g;
    }
    acc += h[(size_t)r * HD + c];
    n += 1.0f;
  }
  if (curg >= 0) {
    atomicAdd(&pooled[curg * HD + c], acc);
    if (c == 0) atomicAdd(&cnt[curg], n);
  }
}

// ---------------------------------------------------------------------------
// Predictor MLP: one block per graph, 128 threads
// ---------------------------------------------------------------------------
__global__ __launch_bounds__(128) void mlp_kernel(const float* __restrict__ pooled,
                                                  const float* __restrict__ cnt,
                                                  const float* __restrict__ W1,
                                                  const float* __restrict__ b1,
                                                  const float* __restrict__ W2,
                                                  const float* __restrict__ b2,
                                                  float* __restrict__ out) {
  __shared__ float sh[128];
  const int g = blockIdx.x;
  const int j = threadIdx.x;
  const float inv = 1.0f / fmaxf(cnt[g], 1.0f);
  float s = b1[j];
  for (int k = 0; k < HD; ++k) s += (pooled[g * HD + k] * inv) * W1[k * 128 + j];
  sh[j] = fmaxf(s, 0.0f) * W2[j];
  __syncthreads();
  for (int off = 64; off > 0; off >>= 1) {
    if (j < off) sh[j] += sh[j + off];
    __syncthreads();
  }
  if (j == 0) out[g] = sh[0] + b2[0];
}

// ---------------------------------------------------------------------------
extern "C" void kernel_launch(void* const* d_in, const int* in_sizes, int n_in,
                              void* d_out, int out_size, void* d_ws, size_t ws_size,
                              hipStream_t stream) {
  constexpr int NN = 100000;   // nodes
  constexpr int NE = 800000;   // edges
  constexpr int IND = 128;     // input dim
  constexpr int NL = 4;        // layers

  const float* x    = (const float*)d_in[0];
  const int*   edge = (const int*)d_in[1];
  const int*   src  = edge;
  const int*   dst  = edge + NE;
  const int*   batch = (const int*)d_in[2];
  const float* Wp   = (const float*)d_in[3];
  const float* bp   = (const float*)d_in[4];
  const float* Wl   = (const float*)d_in[5];   // [4,256,256]
  const float* bl   = (const float*)d_in[6];   // [4,256]
  const float* Wr   = (const float*)d_in[7];   // [4,256,256]
  const float* gamma = (const float*)d_in[8];  // [4,256]
  const float* beta  = (const float*)d_in[9];  // [4,256]
  const float* W1   = (const float*)d_in[10];  // [256,128]
  const float* b1   = (const float*)d_in[11];
  const float* W2   = (const float*)d_in[12];  // [128,1]
  const float* b2   = (const float*)d_in[13];
  float* out = (float*)d_out;

  // workspace layout (floats)
  float* ws      = (float*)d_ws;
  const size_t NH = (size_t)NN * HD;           // 25,600,000
  float* h       = ws;
  float* hn      = ws + NH;
  float* agg     = ws + 2 * NH;
  float* deg     = ws + 3 * NH;                // NN floats (becomes 1/deg)
  float* sums    = deg + NN;                   // 256
  float* sumsq   = sums + HD;                  // 256
  float* scale   = sumsq + HD;                 // 256
  float* shift   = scale + HD;                 // 256
  float* pooled  = shift + HD;                 // 256*256
  float* cnt     = pooled + 256 * HD;          // 256

  const dim3 gemmGrid((NN + 127) / 128, HD / (NT * 16), 1);  // (782, 2)
  const size_t ldsProj  = (size_t)NT * (IND / 4) * 32 * sizeof(v2f);  // 64 KB
  const size_t ldsLayer = (size_t)NT * (HD / 4) * 32 * sizeof(v2f);   // 128 KB

  // degree -> 1/max(deg,1)
  hipMemsetAsync(deg, 0, NN * sizeof(float), stream);
  deg_kernel<<<(NE + 255) / 256, 256, 0, stream>>>(dst, deg, NE);
  invdeg_kernel<<<(NN + 255) / 256, 256, 0, stream>>>(deg, NN);

  // input projection: h = relu(x @ Wp + bp)
  gemm_wmma<true, false><<<gemmGrid, 256, ldsProj, stream>>>(
      x, Wp, nullptr, nullptr, nullptr, bp, h, NN, IND);

  for (int i = 0; i < NL; ++i) {
    hipMemsetAsync(agg, 0, NH * sizeof(float), stream);
    hipMemsetAsync(sums, 0, 2 * HD * sizeof(float), stream);

    scatter_add<<<8192, 256, 0, stream>>>(h, src, dst, agg, NE);

    // hn = (agg * 1/deg) @ Wl[i] + h @ Wr[i] + bl[i]
    gemm_wmma<false, true><<<gemmGrid, 256, ldsLayer, stream>>>(
        agg, Wl + (size_t)i * HD * HD, deg, h, Wr + (size_t)i * HD * HD,
        bl + i * HD, hn, NN, HD);

    bn_stats<<<(NN + 511) / 512, 256, 0, stream>>>(hn, sums, sumsq, NN);
    bn_finalize<<<1, HD, 0, stream>>>(sums, sumsq, gamma + i * HD, beta + i * HD,
                                      scale, shift, 1.0f / (float)NN);
    bn_apply<<<(int)((NH + 255) / 256), 256, 0, stream>>>(hn, scale, shift, h, (int)NH);
  }

  // global mean pool + MLP
  hipMemsetAsync(pooled, 0, (256 * HD + 256) * sizeof(float), stream);
  pool_kernel<<<(NN + 511) / 512, 256, 0, stream>>>(h, batch, pooled, cnt, NN);
  mlp_kernel<<<256, 128, 0, stream>>>(pooled, cnt, W1, b1, W2, b2, out);
}